// GCN_18880676233384
// MI455X (gfx1250) — compile-verified
//
#include <hip/hip_runtime.h>
#include <math.h>

typedef float v2f __attribute__((ext_vector_type(2)));
typedef float v8f __attribute__((ext_vector_type(8)));

#define KDIM 386
#define KMAIN 384   // largest multiple of 4 <= KDIM-? (384+2 tail handled separately)
#define HDIM 256

// ---------------- degree / normalization ----------------
__global__ void gcn_init_deg(float* deg, int N) {
  int i = blockIdx.x * blockDim.x + threadIdx.x;
  if (i < N) deg[i] = 1.0f;
}

__global__ void gcn_accum_deg(const int* __restrict__ dst, float* deg, int E) {
  int e = blockIdx.x * blockDim.x + threadIdx.x;
  if (e < E) atomicAdd(&deg[dst[e]], 1.0f);
}

__global__ void gcn_compute_dis(const float* __restrict__ deg, float* dis, int N) {
  int i = blockIdx.x * blockDim.x + threadIdx.x;
  if (i < N) dis[i] = rsqrtf(deg[i]);
}

__global__ void gcn_edge_norm(const int* __restrict__ src, const int* __restrict__ dst,
                              const float* __restrict__ dis, float* en, int E) {
  int e = blockIdx.x * blockDim.x + threadIdx.x;
  if (e < E) en[e] = dis[src[e]] * dis[dst[e]];
}

// ---------------- GEMM1: t = x @ W1 via fp32 WMMA ----------------
// Each wave computes a 16x32 tile (two 16x16 WMMA accumulators sharing A).
// Block = 256 threads = 8 waves -> one block covers 16 rows x 256 cols.
// K-loop is split: unguarded main loop over k in [0,384), then one explicit
// tail step for K = 384,385 (halfSel==1 lanes supply zeros) so the hot loop
// carries no exec-mask predication.
__global__ __launch_bounds__(256) void gcn_gemm1_wmma(const float* __restrict__ x,
                                                      const float* __restrict__ W1,
                                                      float* __restrict__ t, int M) {
  const int lane = threadIdx.x & 31;
  const int wave = threadIdx.x >> 5;
  const int m0 = blockIdx.x * 16;
  const int n0 = wave * 32;           // 8 waves * 32 cols = 256
  const int halfSel = lane >> 4;      // 0: lanes 0-15, 1: lanes 16-31
  const int l15 = lane & 15;

  v8f c0 = {0.f, 0.f, 0.f, 0.f, 0.f, 0.f, 0.f, 0.f};
  v8f c1 = {0.f, 0.f, 0.f, 0.f, 0.f, 0.f, 0.f, 0.f};

  int row = m0 + l15;
  int rowc = (row < M) ? row : (M - 1);
  const float* arow = x + (size_t)rowc * KDIM;
  const int col0 = n0 + l15;

  // ---- main loop: all lanes fully in-bounds, no predication ----
  for (int k = 0; k < KMAIN; k += 4) {
    const int ka = k + halfSel * 2;   // <= 382+1 < 386 always
    const float2 av = *(const float2*)(arow + ka);  // 8B aligned
    v2f a;  a.x = av.x;  a.y = av.y;
    const float* bp = W1 + (size_t)ka * HDIM + col0;
    v2f b0; b0.x = bp[0];  b0.y = bp[HDIM];
    v2f b1; b1.x = bp[16]; b1.y = bp[HDIM + 16];
    c0 = __builtin_amdgcn_wmma_f32_16x16x4_f32(false, a, false, b0, (short)0, c0,
                                               false, false);
    c1 = __builtin_amdgcn_wmma_f32_16x16x4_f32(false, a, false, b1, (short)0, c1,
                                               false, false);
  }

  // ---- tail: K = 384,385 valid only for halfSel==0 lanes ----
  {
    v2f a  = {0.f, 0.f};
    v2f b0 = {0.f, 0.f};
    v2f b1 = {0.f, 0.f};
    if (halfSel == 0) {
      const float2 av = *(const float2*)(arow + KMAIN);
      a.x = av.x;  a.y = av.y;
      const float* bp = W1 + (size_t)KMAIN * HDIM + col0;
      b0.x = bp[0];  b0.y = bp[HDIM];
      b1.x = bp[16]; b1.y = bp[HDIM + 16];
    }
    c0 = __builtin_amdgcn_wmma_f32_16x16x4_f32(false, a, false, b0, (short)0, c0,
                                               false, false);
    c1 = __builtin_amdgcn_wmma_f32_16x16x4_f32(false, a, false, b1, (short)0, c1,
                                               false, false);
  }

  // C/D layout: VGPR r -> M = r + 8*halfSel, N = lane&15
#pragma unroll
  for (int r = 0; r < 8; ++r) {
    int rr = m0 + r + halfSel * 8;
    if (rr < M) {
      t[(size_t)rr * HDIM + col0] = c0[r];
      t[(size_t)rr * HDIM + col0 + 16] = c1[r];
    }
  }
}

// ---------------- aggregation layer 1 ----------------
__global__ void gcn_agg1_init(const float* __restrict__ t, const float* __restrict__ dis,
                              const float* __restrict__ b1, float* agg, int N) {
  long long i = (long long)blockIdx.x * blockDim.x + threadIdx.x;
  if (i < (long long)N * HDIM) {
    int n = (int)(i >> 8);
    int f = (int)(i & (HDIM - 1));
    float sn = dis[n] * dis[n];
    agg[i] = t[i] * sn + b1[f];
  }
}

// 4 edges per block, 64 lanes per edge, float4 gather + 4 fp32 atomics each.
__global__ __launch_bounds__(256) void gcn_scatter1(const int* __restrict__ src,
                                                    const int* __restrict__ dst,
                                                    const float* __restrict__ en,
                                                    const float* __restrict__ t,
                                                    float* agg, int E) {
  int e = blockIdx.x * 4 + (threadIdx.x >> 6);
  if (e >= E) return;
  int f = (threadIdx.x & 63) * 4;
  int s = src[e], d = dst[e];
  float w = en[e];
  float4 v = *(const float4*)(t + (size_t)s * HDIM + f);
  float* ap = agg + (size_t)d * HDIM + f;
  atomicAdd(ap + 0, v.x * w);
  atomicAdd(ap + 1, v.y * w);
  atomicAdd(ap + 2, v.z * w);
  atomicAdd(ap + 3, v.w * w);
}

// ---------------- GEMM2 (fused ReLU): wave-per-row reduction ----------------
__global__ __launch_bounds__(256) void gcn_gemm2(const float* __restrict__ agg,
                                                 const float* __restrict__ W2,
                                                 float* t2, int N) {
  int row = blockIdx.x * 8 + (threadIdx.x >> 5);
  int lane = threadIdx.x & 31;
  if (row >= N) return;
  const float* ar = agg + (size_t)row * HDIM;
  float s0 = 0.f, s1 = 0.f;
#pragma unroll
  for (int k = lane; k < HDIM; k += 32) {
    float a = ar[k];
    a = a > 0.f ? a : 0.f;  // ReLU
    s0 += a * W2[k * 2 + 0];
    s1 += a * W2[k * 2 + 1];
  }
#pragma unroll
  for (int off = 16; off > 0; off >>= 1) {
    s0 += __shfl_down(s0, off);
    s1 += __shfl_down(s1, off);
  }
  if (lane == 0) {
    t2[row * 2 + 0] = s0;
    t2[row * 2 + 1] = s1;
  }
}

// ---------------- aggregation layer 2 + tanh ----------------
__global__ void gcn_out_init(const float* __restrict__ t2, const float* __restrict__ dis,
                             const float* __restrict__ b2, float* out, int N) {
  int i = blockIdx.x * blockDim.x + threadIdx.x;
  if (i < 2 * N) {
    int n = i >> 1;
    int c = i & 1;
    float sn = dis[n] * dis[n];
    out[i] = t2[i] * sn + b2[c];
  }
}

__global__ void gcn_scatter2(const int* __restrict__ src, const int* __restrict__ dst,
                             const float* __restrict__ en, const float* __restrict__ t2,
                             float* out, int E) {
  int e = blockIdx.x * blockDim.x + threadIdx.x;
  if (e < E) {
    int s = src[e], d = dst[e];
    float w = en[e];
    atomicAdd(&out[d * 2 + 0], t2[s * 2 + 0] * w);
    atomicAdd(&out[d * 2 + 1], t2[s * 2 + 1] * w);
  }
}

__global__ void gcn_tanh(float* out, int n) {
  int i = blockIdx.x * blockDim.x + threadIdx.x;
  if (i < n) out[i] = tanhf(out[i]);
}

// ---------------- launch ----------------
extern "C" void kernel_launch(void* const* d_in, const int* in_sizes, int n_in,
                              void* d_out, int out_size, void* d_ws, size_t ws_size,
                              hipStream_t stream) {
  const float* x  = (const float*)d_in[0];
  const int*   src = (const int*)d_in[1];
  const int*   dst = (const int*)d_in[2];
  const float* W1 = (const float*)d_in[3];
  const float* b1 = (const float*)d_in[4];
  const float* W2 = (const float*)d_in[5];
  const float* b2 = (const float*)d_in[6];
  float* out = (float*)d_out;

  const int N = in_sizes[0] / KDIM;   // 20000
  const int E = in_sizes[1];          // 320000

  float* ws  = (float*)d_ws;
  float* t   = ws;                       // N*256
  float* agg = t   + (size_t)N * HDIM;   // N*256
  float* deg = agg + (size_t)N * HDIM;   // N
  float* dis = deg + N;                  // N
  float* en  = dis + N;                  // E
  float* t2  = en  + E;                  // 2N

  const int TB = 256;
  // normalization
  gcn_init_deg<<<(N + TB - 1) / TB, TB, 0, stream>>>(deg, N);
  gcn_accum_deg<<<(E + TB - 1) / TB, TB, 0, stream>>>(dst, deg, E);
  gcn_compute_dis<<<(N + TB - 1) / TB, TB, 0, stream>>>(deg, dis, N);
  gcn_edge_norm<<<(E + TB - 1) / TB, TB, 0, stream>>>(src, dst, dis, en, E);

  // layer 1
  gcn_gemm1_wmma<<<(N + 15) / 16, 256, 0, stream>>>(x, W1, t, N);
  {
    long long tot = (long long)N * HDIM;
    gcn_agg1_init<<<(int)((tot + TB - 1) / TB), TB, 0, stream>>>(t, dis, b1, agg, N);
  }
  gcn_scatter1<<<(E + 3) / 4, 256, 0, stream>>>(src, dst, en, t, agg, E);

  // layer 2
  gcn_gemm2<<<(N + 7) / 8, 256, 0, stream>>>(agg, W2, t2, N);
  gcn_out_init<<<(2 * N + TB - 1) / TB, TB, 0, stream>>>(t2, dis, b2, out, N);
  gcn_scatter2<<<(E + TB - 1) / TB, TB, 0, stream>>>(src, dst, en, t2, out, E);
  gcn_tanh<<<(2 * N + TB - 1) / TB, TB, 0, stream>>>(out, 2 * N);
}